// Decoder_24816321036414
// MI455X (gfx1250) — compile-verified
//
#include <hip/hip_runtime.h>
#include <hip/hip_bf16.h>
#include <math.h>

#define B_   1024
#define N_   128
#define DH_  128
#define DK_  16
// scale = 1/sqrt(DK) = 0.25 exactly
#define SCALE_ 0.25f

typedef __attribute__((ext_vector_type(2))) float v2f;
typedef __attribute__((ext_vector_type(8))) float v8f;

// ---------------------------------------------------------------------------
// Kernel 1: fused projection GEMM (fp32 WMMA, V_WMMA_F32_16X16X4_F32)
//   Y[r, 0:16]  = x2d[r,:] @ Wka            -> KaG
//   Y[r,16:32]  = x2d[r,:] @ Wqa[128:256]   -> QlG   (vl projection table)
//   Y[r,32:48]  = x2d[r,:] @ Wqa[256:384]   -> QfG   (vf projection table)
// x2d = x viewed as [B*N, DH] = [131072, 128].
// One wave per 16-row strip; 3 accumulators; K looped in chunks of 4.
// Grid: 2048 blocks * 4 waves * 16 rows = 131072 rows, no partial waves
// (EXEC all-ones as WMMA requires).
// ---------------------------------------------------------------------------
__global__ __launch_bounds__(128) void proj_wmma_kernel(
    const float* __restrict__ x,    // [B*N, DH]
    const float* __restrict__ Wqa,  // [384, 16] row-major
    const float* __restrict__ Wka,  // [128, 16] row-major
    float* __restrict__ KaG,        // [B*N, 16]
    float* __restrict__ QlG,        // [B*N, 16]
    float* __restrict__ QfG)        // [B*N, 16]
{
    const int lane = threadIdx.x & 31;
    const int wave = threadIdx.x >> 5;
    const int gw   = blockIdx.x * 4 + wave;   // 0..8191
    const int R    = gw * 16;                 // row base of this 16-row strip

    // ISA VGPR layout for 32-bit 16x4 A (and symmetric 4x16 B):
    //   lane l: m/n = l&15 ; holds K = (l>>4)*2 + {0,1}
    const int mrow = lane & 15;
    const int kb   = (lane >> 4) * 2;
    const int ncol = lane & 15;

    v8f c0 = {}; v8f c1 = {}; v8f c2 = {};
    const float* arow = x + (size_t)(R + mrow) * DH_ + kb;

    #pragma unroll 4
    for (int k0 = 0; k0 < DH_; k0 += 4) {
        v2f a;  a.x = arow[k0];  a.y = arow[k0 + 1];
        const int kk = k0 + kb;
        v2f b0; b0.x = Wka[kk * 16 + ncol];          b0.y = Wka[(kk + 1) * 16 + ncol];
        v2f b1; b1.x = Wqa[(128 + kk) * 16 + ncol];  b1.y = Wqa[(129 + kk) * 16 + ncol];
        v2f b2; b2.x = Wqa[(256 + kk) * 16 + ncol];  b2.y = Wqa[(257 + kk) * 16 + ncol];
        // 8-arg form: (neg_a, A, neg_b, B, c_mod, C, reuse_a, reuse_b)
        c0 = __builtin_amdgcn_wmma_f32_16x16x4_f32(false, a, false, b0, (short)0, c0, false, false);
        c1 = __builtin_amdgcn_wmma_f32_16x16x4_f32(false, a, false, b1, (short)0, c1, false, false);
        c2 = __builtin_amdgcn_wmma_f32_16x16x4_f32(false, a, false, b2, (short)0, c2, false, false);
    }

    // C/D layout: VGPR r holds row M = r + 8*(lane>>4), column N = lane&15.
    #pragma unroll
    for (int r = 0; r < 8; ++r) {
        const int m = r + 8 * (lane >> 4);
        const size_t off = (size_t)(R + m) * DK_ + ncol;
        KaG[off] = c0[r];
        QlG[off] = c1[r];
        QfG[off] = c2[r];
    }
}

// ---------------------------------------------------------------------------
// Kernel 2: sequential greedy decode. One 128-thread block per batch row b.
// Thread t owns candidate n = t (Ka row in registers, mask as a register flag).
// Per step: qa = qm + Ql[prev] + Qf[first] (16 LDS broadcasts), 16-MAC dot,
// wave32 butterfly argmax + logsumexp, O(1) state update. 128 steps.
// ---------------------------------------------------------------------------
__global__ __launch_bounds__(128) void decode_kernel(
    const float* __restrict__ x,     // [B, N, DH]
    const float* __restrict__ vl_p,  // [DH]
    const float* __restrict__ vf_p,  // [DH]
    const float* __restrict__ Wqa,   // [384, 16]
    const float* __restrict__ KaG,   // [B*N, 16]
    const float* __restrict__ QlG,   // [B*N, 16]
    const float* __restrict__ QfG,   // [B*N, 16]
    float* __restrict__ out)         // [B*N] selected (as float), then [B] log_p
{
    __shared__ float Ql_s[N_ * DK_];
    __shared__ float Qf_s[N_ * DK_];
    __shared__ float m_s[DH_];
    __shared__ float qm_s[DK_], qf0_s[DK_], qvl0_s[DK_], qa_s[DK_];
    __shared__ float pmax[4];
    __shared__ int   pidx[4];
    __shared__ float psum[4];
    __shared__ float umax_s;
    __shared__ int   idx_s, prev_s, first_s;

    const int b = blockIdx.x;
    const int t = threadIdx.x;        // 0..127  == candidate index n
    const int lane = t & 31;
    const int wv   = t >> 5;

    // Per-thread Ka row -> registers; lookup tables -> LDS.
    float kat[DK_];
    {
        const size_t base = ((size_t)b * N_ + t) * DK_;
        #pragma unroll
        for (int k = 0; k < DK_; ++k) kat[k] = KaG[base + k];
        #pragma unroll
        for (int k = 0; k < DK_; ++k) {
            Ql_s[t * DK_ + k] = QlG[base + k];
            Qf_s[t * DK_ + k] = QfG[base + k];
        }
    }
    // m = x[b].mean(axis=0): thread t reduces column d = t (coalesced across t).
    {
        float s = 0.f;
        const float* xb = x + (size_t)b * N_ * DH_ + t;
        for (int n = 0; n < N_; ++n) s += xb[(size_t)n * DH_];
        m_s[t] = s * (1.0f / N_);
    }
    __syncthreads();

    // Loop-invariant pieces of qa (computed once per block; 128 MACs/thread).
    if (t < 16) {
        float s = 0.f;
        for (int d = 0; d < DH_; ++d) s += m_s[d] * Wqa[d * 16 + t];
        qm_s[t] = s;
    } else if (t < 32) {
        const int k = t - 16; float s = 0.f;
        for (int d = 0; d < DH_; ++d) s += vf_p[d] * Wqa[(256 + d) * 16 + k];
        qf0_s[k] = s;
    } else if (t < 48) {
        const int k = t - 32; float s = 0.f;
        for (int d = 0; d < DH_; ++d) s += vl_p[d] * Wqa[(128 + d) * 16 + k];
        qvl0_s[k] = s;
    }
    if (t == 0) { prev_s = 0; first_s = 0; }

    bool visited = false;
    float log_p = 0.f;
    __syncthreads();

    for (int i = 0; i < N_; ++i) {
        // qa = qm + proj_l(vl) + proj_f(vf)
        if (t < 16) {
            const float ql = (i == 0) ? qvl0_s[t] : Ql_s[prev_s  * DK_ + t];
            const float qf = (i == 0) ? qf0_s[t]  : Qf_s[first_s * DK_ + t];
            qa_s[t] = qm_s[t] + ql + qf;
        }
        __syncthreads();

        // u[n] = scale * qa . Ka[b,n]  (masked candidates -> -inf)
        float u;
        if (visited) {
            u = -__builtin_inff();
        } else {
            float s = 0.f;
            #pragma unroll
            for (int k = 0; k < DK_; ++k) s += qa_s[k] * kat[k];
            u = s * SCALE_;
        }

        // argmax (first-index tie-break, matching jnp.argmax): wave32 butterfly.
        float mu = u; int mi = t;
        #pragma unroll
        for (int off = 16; off; off >>= 1) {
            const float ou = __shfl_xor(mu, off, 32);
            const int   oi = __shfl_xor(mi, off, 32);
            if (ou > mu || (ou == mu && oi < mi)) { mu = ou; mi = oi; }
        }
        if (lane == 0) { pmax[wv] = mu; pidx[wv] = mi; }
        __syncthreads();
        if (t == 0) {
            float bm = pmax[0]; int bi = pidx[0];
            #pragma unroll
            for (int w = 1; w < 4; ++w)
                if (pmax[w] > bm || (pmax[w] == bm && pidx[w] < bi)) { bm = pmax[w]; bi = pidx[w]; }
            umax_s = bm; idx_s = bi;
        }
        __syncthreads();

        // max(log_softmax(u)) = -log( sum exp(u - umax) )   (exp(-inf)=0 for masked)
        float e = expf(u - umax_s);
        #pragma unroll
        for (int off = 16; off; off >>= 1) e += __shfl_xor(e, off, 32);
        if (lane == 0) psum[wv] = e;
        __syncthreads();

        if (t == idx_s) visited = true;   // mask[b, idx] = inf
        if (t == 0) {
            const float S = psum[0] + psum[1] + psum[2] + psum[3];
            log_p -= logf(S);
            out[(size_t)b * N_ + i] = (float)idx_s;   // selected[b, i]
            prev_s = idx_s;                            // vl <- x[b, idx]
            if (i == 0) first_s = idx_s;               // vf frozen at step 0
        }
        __syncthreads();
    }

    if (t == 0) out[(size_t)B_ * N_ + b] = log_p;
}

extern "C" void kernel_launch(void* const* d_in, const int* in_sizes, int n_in,
                              void* d_out, int out_size, void* d_ws, size_t ws_size,
                              hipStream_t stream) {
    (void)in_sizes; (void)n_in; (void)out_size; (void)ws_size;
    const float* x    = (const float*)d_in[0];
    const float* vl_p = (const float*)d_in[1];
    const float* vf_p = (const float*)d_in[2];
    // d_in[3..6] = Wq, Wk, Wv, Wo: dead code in the reference (h is discarded).
    const float* Wqa  = (const float*)d_in[7];
    const float* Wka  = (const float*)d_in[8];
    float* out = (float*)d_out;

    // Workspace: three [B*N, 16] f32 tables (8 MB each, 24 MB total).
    const size_t tbl = (size_t)B_ * N_ * DK_;
    float* KaG = (float*)d_ws;
    float* QlG = KaG + tbl;
    float* QfG = QlG + tbl;

    // 131072 rows / (16 rows/wave * 4 waves/block) = 2048 blocks.
    proj_wmma_kernel<<<2048, 128, 0, stream>>>(x, Wqa, Wka, KaG, QlG, QfG);
    decode_kernel<<<B_, 128, 0, stream>>>(x, vl_p, vf_p, Wqa, KaG, QlG, QfG, out);
}